// TLSTM_base_36404142801119
// MI455X (gfx1250) — compile-verified
//
#include <hip/hip_runtime.h>
#include <hip/hip_bf16.h>
#include <math.h>

// ---------------------------------------------------------------------------
// TLSTM forward for MI455X (gfx1250), fp32 path via V_WMMA_F32_16X16X4_F32.
// B=64, S=128, C=32, D=512, V=20000.
//
// Pipeline (all on `stream`):
//   1. embed_sum:   x[8192,512]  = sum_c emb[idx]          (L2-resident gather)
//   2. timefeat:    tf[8192,512] = (1 - tanh((t*sel)^2)) @ time_w^T + time_b
//   3. ux_gemm:     ux[8192,2048]= x @ U^T + (U_b + Wall_b)  [fp32 WMMA]
//   4. recurrence:  persistent kernel, 128 sequential steps, grid-wide
//                   atomic barrier; per step 640 16x16 tiles via fp32 WMMA
//   5. out_gemm:    [64,2] = hmax @ out_w^T + out_b
// ---------------------------------------------------------------------------

typedef __attribute__((ext_vector_type(2))) float v2f;
typedef __attribute__((ext_vector_type(8))) float v8f;

#define Bdim 64
#define Sdim 128
#define Cdim 32
#define Ddim 512

// workspace offsets (in floats)
#define OFF_X      0u
#define OFF_TF     4194304u            // 8192*512
#define OFF_UX     8388608u            // + 8192*512
#define OFF_H      25165824u           // + 8192*2048
#define OFF_C      25198592u           // + 64*512
#define OFF_CADJ   25231360u
#define OFF_GATES  25264128u
#define OFF_HMAX   25395200u           // + 64*2048
#define OFF_SYNC   25427968u           // + 64*512

// ---------------------------------------------------------------------------
// 1. embedding gather + sum over 32 codes per visit
// ---------------------------------------------------------------------------
__global__ void __launch_bounds__(256) embed_sum_kernel(
    const float* __restrict__ emb, const int* __restrict__ seqs,
    float* __restrict__ x)
{
    const int row = blockIdx.x;                 // b*S + s
    __shared__ int idx[Cdim];
    if (threadIdx.x < Cdim) idx[threadIdx.x] = seqs[row * Cdim + threadIdx.x];
    __syncthreads();
    for (int d = threadIdx.x; d < Ddim; d += 256) {
        float acc = 0.0f;
#pragma unroll 8
        for (int c = 0; c < Cdim; ++c)
            acc += emb[idx[c] * Ddim + d];
        x[row * Ddim + d] = acc;
    }
}

// ---------------------------------------------------------------------------
// 2. time feature: g[k] = 1 - tanh((t/180*sel_w[k]+sel_b[k])^2), tf = g@time_w^T+b
// ---------------------------------------------------------------------------
__global__ void __launch_bounds__(256) timefeat_kernel(
    const float* __restrict__ sel_w, const float* __restrict__ sel_b,
    const float* __restrict__ time_w, const float* __restrict__ time_b,
    const float* __restrict__ tstep, float* __restrict__ tf)
{
    const int row = blockIdx.x;                 // b*S + s
    __shared__ float g[64];
    const float tval = tstep[row] * (1.0f / 180.0f);
    if (threadIdx.x < 64) {
        float u = tval * sel_w[threadIdx.x] + sel_b[threadIdx.x];
        g[threadIdx.x] = 1.0f - tanhf(u * u);
    }
    __syncthreads();
    for (int d = threadIdx.x; d < Ddim; d += 256) {
        float acc = time_b[d];
        const float* wrow = time_w + d * 64;
#pragma unroll
        for (int k = 0; k < 64; ++k) acc += g[k] * wrow[k];
        tf[row * Ddim + d] = acc;
    }
}

// ---------------------------------------------------------------------------
// 3. ux = x @ U^T + (U_b + Wall_b)   [8192,512]x[512,2048]
//    grid (32, 64), 8 waves/block; each wave: 16 rows x 64 cols (4 tiles),
//    K-loop of 128 steps of V_WMMA_F32_16X16X4_F32 with A-fragment reuse.
// ---------------------------------------------------------------------------
__global__ void __launch_bounds__(256) ux_gemm_kernel(
    const float* __restrict__ x, const float* __restrict__ U,
    const float* __restrict__ Ub, const float* __restrict__ Wb,
    float* __restrict__ ux)
{
    const unsigned lane = threadIdx.x & 31u;
    const unsigned wv   = threadIdx.x >> 5;
    const unsigned lm   = lane & 15u;           // row/col within 16-tile
    const unsigned kb   = (lane >> 4) << 1;     // K sub-offset {0,2}
    const unsigned rowoff = (lane >> 4) << 3;   // +8 rows for upper lanes

    const int m0 = blockIdx.y * 128 + wv * 16;
    const int n0 = blockIdx.x * 64;

    const float* ap = x + (m0 + lm) * Ddim;
    const float* bp = U + (n0 + lm) * Ddim;

    v8f a0 = {}, a1 = {}, a2 = {}, a3 = {};
#pragma unroll 4
    for (int k = 0; k < Ddim; k += 4) {
        v2f a  = *(const v2f*)(ap + k + kb);
        v2f b0 = *(const v2f*)(bp + k + kb);
        v2f b1 = *(const v2f*)(bp + 16 * Ddim + k + kb);
        v2f b2 = *(const v2f*)(bp + 32 * Ddim + k + kb);
        v2f b3 = *(const v2f*)(bp + 48 * Ddim + k + kb);
        a0 = __builtin_amdgcn_wmma_f32_16x16x4_f32(false, a, false, b0, (short)0, a0, false, false);
        a1 = __builtin_amdgcn_wmma_f32_16x16x4_f32(false, a, false, b1, (short)0, a1, false, false);
        a2 = __builtin_amdgcn_wmma_f32_16x16x4_f32(false, a, false, b2, (short)0, a2, false, false);
        a3 = __builtin_amdgcn_wmma_f32_16x16x4_f32(false, a, false, b3, (short)0, a3, false, false);
    }

#pragma unroll
    for (int j = 0; j < 4; ++j) {
        v8f acc = (j == 0) ? a0 : (j == 1) ? a1 : (j == 2) ? a2 : a3;
        const int n = n0 + j * 16 + (int)lm;
        const float bias = Ub[n] + Wb[n];       // fold both biases here
#pragma unroll
        for (int r = 0; r < 8; ++r) {
            const int m = m0 + r + (int)rowoff;
            ux[(unsigned)m * 2048u + (unsigned)n] = acc[r] + bias;
        }
    }
}

// ---------------------------------------------------------------------------
// device-wide two-phase barrier (all 40 blocks resident)
// ---------------------------------------------------------------------------
__device__ __forceinline__ void grid_sync(unsigned* bar, unsigned* gen, unsigned nblk)
{
    __threadfence();
    __syncthreads();
    if (threadIdx.x == 0) {
        unsigned g = atomicAdd(gen, 0u);
        if (atomicAdd(bar, 1u) == nblk - 1u) {
            atomicExch(bar, 0u);
            __threadfence();
            atomicAdd(gen, 1u);
        } else {
            while (atomicAdd(gen, 0u) == g) { __builtin_amdgcn_s_sleep(1); }
        }
        __threadfence();
    }
    __syncthreads();
}

// ---------------------------------------------------------------------------
// 4. persistent recurrence: 40 blocks x 8 waves = 320 waves.
//    Per step: 512 gate tiles (h @ W_all^T) + 128 decay tiles (c @ W_d^T)
//    -> exactly 2 tiles per wave; grid barrier; elementwise cell update;
//    grid barrier.
// ---------------------------------------------------------------------------
__global__ void __launch_bounds__(256) recurrence_kernel(
    const float* __restrict__ W_all, const float* __restrict__ W_d,
    const float* __restrict__ W_d_b,
    const float* __restrict__ tf, const float* __restrict__ ux,
    float* __restrict__ h, float* __restrict__ c,
    float* __restrict__ cadj, float* __restrict__ gates,
    float* __restrict__ hmax, unsigned* __restrict__ sync)
{
    const unsigned lane   = threadIdx.x & 31u;
    const unsigned lm     = lane & 15u;
    const unsigned kb     = (lane >> 4) << 1;
    const unsigned rowoff = (lane >> 4) << 3;
    const unsigned gtid   = blockIdx.x * blockDim.x + threadIdx.x;
    const unsigned nthr   = gridDim.x * blockDim.x;     // 10240
    const unsigned wv     = gtid >> 5;                  // 0..319

    for (int s = 0; s < Sdim; ++s) {
        // -------- phase 1: the two GEMMs off old h / old c ----------------
        for (unsigned t = wv; t < 640u; t += 320u) {
            if (t < 512u) {
                // gate GEMM: outs[64,2048] = h @ W_all^T (+ux: bias folded in)
                const unsigned m0 = (t & 3u) * 16u, n0 = (t >> 2) * 16u;
                const float* ap = h + (m0 + lm) * Ddim;
                const float* bp = W_all + (n0 + lm) * Ddim;
                v8f acc = {};
#pragma unroll 4
                for (int k = 0; k < Ddim; k += 4) {
                    v2f a = *(const v2f*)(ap + k + kb);
                    v2f b = *(const v2f*)(bp + k + kb);
                    acc = __builtin_amdgcn_wmma_f32_16x16x4_f32(false, a, false, b, (short)0, acc, false, false);
                }
                const unsigned n = n0 + lm;
#pragma unroll
                for (int r = 0; r < 8; ++r) {
                    const unsigned m = m0 + (unsigned)r + rowoff;      // batch b
                    float v = acc[r] + ux[(m * 128u + (unsigned)s) * 2048u + n];
                    gates[m * 2048u + n] = 1.0f / (1.0f + expf(-v));   // sigmoid
                }
            } else {
                // decay GEMM: c_s1 = tanh(c @ W_d^T + b); c_adj fused here
                const unsigned t2 = t - 512u;
                const unsigned m0 = (t2 & 3u) * 16u, n0 = (t2 >> 2) * 16u;
                const float* ap = c + (m0 + lm) * Ddim;
                const float* bp = W_d + (n0 + lm) * Ddim;
                v8f acc = {};
#pragma unroll 4
                for (int k = 0; k < Ddim; k += 4) {
                    v2f a = *(const v2f*)(ap + k + kb);
                    v2f b = *(const v2f*)(bp + k + kb);
                    acc = __builtin_amdgcn_wmma_f32_16x16x4_f32(false, a, false, b, (short)0, acc, false, false);
                }
                const unsigned n = n0 + lm;
                const float bias = W_d_b[n];
#pragma unroll
                for (int r = 0; r < 8; ++r) {
                    const unsigned m = m0 + (unsigned)r + rowoff;
                    float cs1  = tanhf(acc[r] + bias);
                    float cold = c[m * 512u + n];
                    float tfv  = tf[(m * 128u + (unsigned)s) * 512u + n];
                    cadj[m * 512u + n] = cold - cs1 + cs1 * tfv;
                }
            }
        }
        grid_sync(sync, sync + 1, gridDim.x);

        // -------- phase 2: elementwise cell/hidden update + running max ---
        for (unsigned e = gtid; e < 32768u; e += nthr) {
            const unsigned b = e >> 9, n = e & 511u;
            const float* g = gates + b * 2048u;
            float f  = g[n];
            float i  = g[512u + n];
            float o  = g[1024u + n];
            float ct = g[1536u + n];
            float cn = f * cadj[e] + i * ct;
            float hn = o * tanhf(cn);
            c[e] = cn;
            h[e] = hn;
            hmax[e] = fmaxf(hmax[e], hn);
        }
        grid_sync(sync, sync + 1, gridDim.x);
    }
}

// ---------------------------------------------------------------------------
// init hmax to -inf (h in (-1,1), so -3.4e38 is safely below)
// ---------------------------------------------------------------------------
__global__ void __launch_bounds__(256) init_hmax_kernel(float* __restrict__ hmax)
{
    unsigned e = blockIdx.x * 256u + threadIdx.x;
    if (e < 32768u) hmax[e] = -3.402823466e38f;
}

// ---------------------------------------------------------------------------
// 5. out[64,2] = hmax @ out_w^T + out_b
// ---------------------------------------------------------------------------
__global__ void __launch_bounds__(128) out_gemm_kernel(
    const float* __restrict__ hmax, const float* __restrict__ out_w,
    const float* __restrict__ out_b, float* __restrict__ out)
{
    const int t = threadIdx.x;                  // 0..127
    const int b = t >> 1, j = t & 1;
    float acc = out_b[j];
    const float* hp = hmax + b * Ddim;
    const float* wp = out_w + j * Ddim;
#pragma unroll 8
    for (int d = 0; d < Ddim; ++d) acc += hp[d] * wp[d];
    out[b * 2 + j] = acc;
}

// ---------------------------------------------------------------------------
extern "C" void kernel_launch(void* const* d_in, const int* in_sizes, int n_in,
                              void* d_out, int out_size, void* d_ws, size_t ws_size,
                              hipStream_t stream)
{
    const float* emb     = (const float*)d_in[0];
    const float* W_all_w = (const float*)d_in[1];
    const float* W_all_b = (const float*)d_in[2];
    const float* U_all_w = (const float*)d_in[3];
    const float* U_all_b = (const float*)d_in[4];
    const float* W_d_w   = (const float*)d_in[5];
    const float* W_d_b   = (const float*)d_in[6];
    const float* sel_w   = (const float*)d_in[7];
    const float* sel_b   = (const float*)d_in[8];
    const float* time_w  = (const float*)d_in[9];
    const float* time_b  = (const float*)d_in[10];
    const float* out_w   = (const float*)d_in[11];
    const float* out_b   = (const float*)d_in[12];
    const float* tstep   = (const float*)d_in[13];
    const int*   seqs    = (const int*)d_in[14];
    // mask/lengths/codemask/label unused by the reference output

    float* ws    = (float*)d_ws;
    float* x     = ws + OFF_X;
    float* tf    = ws + OFF_TF;
    float* ux    = ws + OFF_UX;
    float* h     = ws + OFF_H;
    float* c     = ws + OFF_C;
    float* cadj  = ws + OFF_CADJ;
    float* gates = ws + OFF_GATES;
    float* hmax  = ws + OFF_HMAX;
    unsigned* sync = (unsigned*)(ws + OFF_SYNC);

    // zero h, c (contiguous) and barrier state — inside the captured graph
    hipMemsetAsync(h, 0, 2u * 32768u * sizeof(float), stream);
    hipMemsetAsync(sync, 0, 16, stream);

    embed_sum_kernel<<<Bdim * Sdim, 256, 0, stream>>>(emb, seqs, x);
    timefeat_kernel<<<Bdim * Sdim, 256, 0, stream>>>(sel_w, sel_b, time_w, time_b, tstep, tf);
    init_hmax_kernel<<<128, 256, 0, stream>>>(hmax);
    ux_gemm_kernel<<<dim3(32, 64), 256, 0, stream>>>(x, U_all_w, U_all_b, W_all_b, ux);
    recurrence_kernel<<<40, 256, 0, stream>>>(W_all_w, W_d_w, W_d_b, tf, ux,
                                              h, c, cadj, gates, hmax, sync);
    out_gemm_kernel<<<1, 128, 0, stream>>>(hmax, out_w, out_b, (float*)d_out);
}